// HybridQNN_71176198029596
// MI455X (gfx1250) — compile-verified
//
#include <hip/hip_runtime.h>
#include <math.h>

// ---------------------------------------------------------------------------
// HybridQNN on MI455X (gfx1250, wave32)
//
//   q_in = tanh(x @ enc_w^T + enc_b) * pi/2          (65536x512)x(512x4)
//   circuit(q_in) with batch-shared weights
//
// The circuit after the RY-embedding is a FIXED real 16x16 matrix U
// (RY layers + CNOT ring with shared q_params). The embedded state is a
// real product state v (per-qubit [cos(a/2+pi/4), sin(a/2+pi/4)]).
// Outputs: z_q = sum_i sign_q(i) * (U v)_i^2.
//
// Bandwidth bound: 128 MB of x @ 23.3 TB/s ~= 5.5us floor. GEMM done with
// V_WMMA_F32_16X16X4_F32 (full fp32 precision, N padded 4->16), with two
// independent accumulators to break the WMMA->WMMA RAW chain on D.
// ---------------------------------------------------------------------------

typedef __attribute__((ext_vector_type(2))) float v2f;
typedef __attribute__((ext_vector_type(8))) float v8f;

#define NQ        4
#define QDEPTH    6
#define FDIM      512
#define TILE_M    16
#define WAVES_PB  8
#define ROWS_PB   (TILE_M * WAVES_PB)   // 128 rows per block

// ---------------------------------------------------------------------------
// Prep kernel (1 block): build padded/transposed encoder matrix encT[512][16]
// and the fixed 16x16 circuit matrix U from q_params.
// Flat-state convention (matches jnp.reshape row-major): qubit q <-> bit (3-q).
// ---------------------------------------------------------------------------
__global__ void qnn_prep(const float* __restrict__ enc_w,     // [4][512]
                         const float* __restrict__ q_params,  // [4][6]
                         float* __restrict__ encT,            // [512][16]
                         float* __restrict__ U)               // [16][16] row-major
{
    const int tid = threadIdx.x;

    // encT[k][n] = (n < 4) ? enc_w[n][k] : 0
    for (int idx = tid; idx < FDIM * 16; idx += blockDim.x) {
        const int k = idx >> 4;
        const int n = idx & 15;
        encT[idx] = (n < NQ) ? enc_w[n * FDIM + k] : 0.0f;
    }

    // Column j of U = (gates) e_j ; one thread per column.
    if (tid < 16) {
        float s[16];
#pragma unroll
        for (int i = 0; i < 16; ++i) s[i] = (i == tid) ? 1.0f : 0.0f;

        for (int d = 0; d < QDEPTH; ++d) {
            // RY layer (all on distinct qubits -> order free)
            for (int q = 0; q < NQ; ++q) {
                const float th = 0.5f * q_params[q * QDEPTH + d];
                const float c  = cosf(th);
                const float sn = sinf(th);
                const int   m  = 1 << (3 - q);
#pragma unroll
                for (int i = 0; i < 16; ++i) {
                    if (i & m) continue;
                    const float a = s[i], b = s[i | m];
                    s[i]     = c * a - sn * b;
                    s[i | m] = sn * a + c * b;
                }
            }
            // CNOT ring: control q, target (q+1)%4, applied in order
            for (int q = 0; q < NQ; ++q) {
                const int mc = 1 << (3 - q);
                const int mt = 1 << (3 - ((q + 1) & 3));
#pragma unroll
                for (int i = 0; i < 16; ++i) {
                    if ((i & mc) && !(i & mt)) {
                        const float t = s[i];
                        s[i]      = s[i | mt];
                        s[i | mt] = t;
                    }
                }
            }
        }
#pragma unroll
        for (int i = 0; i < 16; ++i) U[i * 16 + tid] = s[i];
    }
}

// ---------------------------------------------------------------------------
// Main kernel: 256 threads = 8 waves, each wave owns a 16-row tile.
// ---------------------------------------------------------------------------
__global__ void __launch_bounds__(256)
qnn_main(const float* __restrict__ x,     // [B][512]
         const float* __restrict__ encT,  // [512][16]
         const float* __restrict__ enc_b, // [4]
         const float* __restrict__ U,     // [16][16]
         float* __restrict__ out)         // [B][4]
{
    __shared__ float shB[FDIM * 16];                  // 32 KB: B fragments
    __shared__ float shU[256];                        //  1 KB: circuit matrix
    __shared__ float shAng[WAVES_PB][TILE_M][NQ];     //  2 KB: angle transpose

    const int tid  = threadIdx.x;
    const int wave = tid >> 5;
    const int lane = tid & 31;
    const int half = lane >> 4;   // 0: K=0,1  1: K=2,3  (A-frag ISA layout)
    const int col  = lane & 15;   // M for A, N for B/C/D

    // Stage shared operands (coalesced), reused by all 8 waves.
    for (int i = tid; i < FDIM * 16; i += 256) shB[i] = encT[i];
    shU[tid] = U[tid];
    __syncthreads();

    const int    rowBase = (blockIdx.x * WAVES_PB + wave) * TILE_M;
    const float* aptr    = x + (size_t)(rowBase + col) * FDIM + 2 * half;

    // ---- GEMM: acc[16x16] += A[16x4] * B[4x16] over K=512 -----------------
    // Two independent accumulators (even/odd k-steps) break the serial
    // WMMA->WMMA RAW dependency on D (ISA 7.12.1 hazard chain).
    v8f acc0 = {0.f, 0.f, 0.f, 0.f, 0.f, 0.f, 0.f, 0.f};
    v8f acc1 = {0.f, 0.f, 0.f, 0.f, 0.f, 0.f, 0.f, 0.f};
    for (int k0 = 0; k0 < FDIM; k0 += 8) {
        // A: lane holds x[rowBase+col][k + 2*half + {0,1}]
        v2f a0 = *(const v2f*)(aptr + k0);
        v2f a1 = *(const v2f*)(aptr + k0 + 4);
        // B: VGPR0 <-> K = k+2*half, VGPR1 <-> K = k+2*half+1, N = col
        const int kk0 = (k0 + 2 * half) << 4;
        const int kk1 = (k0 + 4 + 2 * half) << 4;
        v2f b0, b1;
        b0.x = shB[kk0 + col];
        b0.y = shB[kk0 + 16 + col];
        b1.x = shB[kk1 + col];
        b1.y = shB[kk1 + 16 + col];
        acc0 = __builtin_amdgcn_wmma_f32_16x16x4_f32(
            false, a0, false, b0, (short)0, acc0, false, false);
        acc1 = __builtin_amdgcn_wmma_f32_16x16x4_f32(
            false, a1, false, b1, (short)0, acc1, false, false);
    }
    v8f acc = acc0 + acc1;

    // ---- bias + tanh + scale; transpose rows through LDS ------------------
    const float HALF_PI = 1.5707963267948966f;
    if (col < NQ) {
        const float bias = enc_b[col];
#pragma unroll
        for (int j = 0; j < 8; ++j) {
            // C/D layout: lane half 0 -> rows 0..7, half 1 -> rows 8..15
            shAng[wave][half * 8 + j][col] = tanhf(acc[j] + bias) * HALF_PI;
        }
    }
    __syncthreads();

    // ---- 4-qubit circuit per row (lanes 0..15 of each wave) ---------------
    if (lane < TILE_M) {
        const int r = lane;
        float cq[NQ], sq[NQ];
#pragma unroll
        for (int q = 0; q < NQ; ++q) {
            const float phi = 0.5f * shAng[wave][r][q] + 0.7853981633974483f;
            cq[q] = cosf(phi);
            sq[q] = sinf(phi);
        }
        // Embedded product state, qubit q <-> bit (3-q)
        float v[16];
#pragma unroll
        for (int i = 0; i < 16; ++i) {
            float p = ((i >> 3) & 1) ? sq[0] : cq[0];
            p *= ((i >> 2) & 1) ? sq[1] : cq[1];
            p *= ((i >> 1) & 1) ? sq[2] : cq[2];
            p *= ( i       & 1) ? sq[3] : cq[3];
            v[i] = p;
        }
        // w = U v ; p2 = w*w
        float p2[16];
#pragma unroll
        for (int i = 0; i < 16; ++i) {
            float a = 0.f;
#pragma unroll
            for (int j = 0; j < 16; ++j) a = fmaf(shU[i * 16 + j], v[j], a);
            p2[i] = a * a;
        }
        // z_q = sum_i sign * p2[i]
        float z[NQ];
#pragma unroll
        for (int q = 0; q < NQ; ++q) {
            const int m = 1 << (3 - q);
            float zz = 0.f;
#pragma unroll
            for (int i = 0; i < 16; ++i) zz += (i & m) ? -p2[i] : p2[i];
            z[q] = zz;
        }
        *(float4*)(out + (size_t)(rowBase + r) * NQ) =
            make_float4(z[0], z[1], z[2], z[3]);
    }
}

// ---------------------------------------------------------------------------
extern "C" void kernel_launch(void* const* d_in, const int* in_sizes, int n_in,
                              void* d_out, int out_size, void* d_ws, size_t ws_size,
                              hipStream_t stream)
{
    const float* x        = (const float*)d_in[0];  // (65536, 512) f32
    const float* enc_w    = (const float*)d_in[1];  // (4, 512) f32
    const float* enc_b    = (const float*)d_in[2];  // (4,) f32
    const float* q_params = (const float*)d_in[3];  // (4, 6) f32

    float* encT = (float*)d_ws;          // 512*16 floats = 32 KB
    float* U    = encT + FDIM * 16;      // 16*16 floats  =  1 KB
    float* out  = (float*)d_out;         // (65536, 4) f32

    qnn_prep<<<1, 256, 0, stream>>>(enc_w, q_params, encT, U);

    const int B      = in_sizes[0] / FDIM;               // 65536
    const int blocks = (B + ROWS_PB - 1) / ROWS_PB;      // 512
    qnn_main<<<blocks, 256, 0, stream>>>(x, encT, enc_b, U, out);
}